// CapsNet_9955734192312
// MI455X (gfx1250) — compile-verified
//
#include <hip/hip_runtime.h>
#include <hip/hip_bf16.h>
#include <math.h>

typedef float v2f __attribute__((ext_vector_type(2)));
typedef float v8f __attribute__((ext_vector_type(8)));

#define NBATCH 8
#define HWDIM 14
#define NPOS (NBATCH*HWDIM*HWDIM)   /* 1568 */
#define ICAPS 32
#define PP 16
#define EPSR 1e-6f

// ---------------- conv1: 5x5 stride 2 pad 2, relu ----------------
__global__ void caps_conv1_kernel(const float* __restrict__ x, const float* __restrict__ w,
                                  const float* __restrict__ b, float* __restrict__ y) {
  int t = blockIdx.x * blockDim.x + threadIdx.x;
  if (t >= NBATCH * 32 * HWDIM * HWDIM) return;
  int j = t % 14; int i = (t / 14) % 14; int a = (t / 196) % 32; int n = t / (196 * 32);
  float acc = b[a];
  const float* xp = x + n * 28 * 28;
  const float* wp = w + a * 25;
  int r0 = 2 * i - 2, c0 = 2 * j - 2;
  for (int u = 0; u < 5; ++u) {
    int r = r0 + u;
    if (r < 0 || r >= 28) continue;
    for (int v = 0; v < 5; ++v) {
      int c = c0 + v;
      if (c < 0 || c >= 28) continue;
      acc += xp[r * 28 + c] * wp[u * 5 + v];
    }
  }
  y[t] = fmaxf(acc, 0.f);
}

// ------- pconv (1x1) fused with the torch-style raw reshape into -------
// pose0[pos][i][k] (pos = ((n*14+h)*14+w)) and act0[pos][i]
__global__ void caps_pconv_kernel(const float* __restrict__ y, const float* __restrict__ pw,
                                  const float* __restrict__ pb,
                                  float* __restrict__ pose0, float* __restrict__ act0) {
  const int NPOSE = NPOS * ICAPS * PP;   // 802816
  int t = blockIdx.x * blockDim.x + threadIdx.x;
  if (t >= NPOSE + NPOS * ICAPS) return;
  int n, c, si, sj, dst;
  float* outp;
  if (t < NPOSE) {
    int pos = t / (ICAPS * PP);
    int rem = t % (ICAPS * PP);          // i*16 + k
    n = pos / 196;
    int hw = pos % 196;
    int g = hw * (ICAPS * PP) + rem;     // flat over (h,w,i,k)
    c = 32 + g / 196;
    int s = g % 196;
    si = s / 14; sj = s % 14;
    outp = pose0; dst = t;
  } else {
    int u = t - NPOSE;
    int pos = u / ICAPS; int bc = u % ICAPS;
    n = pos / 196;
    int hw = pos % 196;
    int f = hw * ICAPS + bc;
    c = f / 196;
    int s = f % 196;
    si = s / 14; sj = s % 14;
    outp = act0; dst = u;
  }
  const float* yp = y + (size_t)n * 32 * 196 + si * 14 + sj;
  const float* wp = pw + (size_t)c * 32;
  float acc = pb[c];
  for (int a = 0; a < 32; ++a) acc += yp[a * 196] * wp[a];
  outp[dst] = acc;
}

// ---------------- one EM-routing capsule layer ----------------
// block = 128 threads (4 wave32), 4 spatial positions per block.
__global__ __launch_bounds__(128)
void caps_layer_kernel(const float* __restrict__ pose_in, const float* __restrict__ act_in,
                       const float* __restrict__ T, const float* __restrict__ beta_v,
                       const float* __restrict__ beta_a,
                       float* __restrict__ pose_out, float* __restrict__ act_out, int Oc) {
  __shared__ float poseS[4 * 32 * 16];
  __shared__ float meansS[4 * 32 * 16];
  __shared__ float varS[4 * 32 * 16];
  __shared__ float actS[4 * 32];
  __shared__ float cinvS[4 * 32];
  __shared__ float cbS[4 * 32];
  __shared__ float GS[4 * 32];
  __shared__ float HS[4 * 32];
  __shared__ float asumS[4];

  const int tid = threadIdx.x;
  const int lane = tid & 31;
  const int wid = tid >> 5;
  const int pos0 = blockIdx.x * 4;
  const int Ntiles = (Oc + 3) >> 2;

  // stage pose tile (4 pos x 32 caps x 16) and act
  for (int e = tid; e < 4 * 32 * 16; e += 128) {
    int sp = e >> 9;
    poseS[e] = pose_in[(size_t)(pos0 + sp) * 512 + (e & 511)];
  }
  {
    int sp = tid >> 5;
    actS[tid] = act_in[(size_t)(pos0 + sp) * 32 + (tid & 31)];
  }
  __syncthreads();

  const int hi = lane >> 4;        // which half-wave
  const int nn = lane & 15;
  const int spA = nn >> 2, pA = nn & 3;   // A-operand: row m = 4*sp + p
  const int kq0 = hi * 2;                  // K pair for this half
  const int ol = nn >> 2, rr = nn & 3;     // B/D column: n = 4*ol + rr

  // -------- pass 1: S1 = sum_i votes, var = S2 + (I-2)*S1^2 --------
  for (int ot = wid; ot < Ntiles; ot += 4) {
    int obase = ot << 2;
    int o = obase + ol;
    int oclamp = (o < Oc) ? o : 0;          // branch-free padded-o handling
    float msk = (o < Oc) ? 1.f : 0.f;
    const float* tbase = T + ((size_t)oclamp * 4 + kq0) * 4 + rr;
    float s1[8], s2[8];
    for (int j = 0; j < 8; ++j) { s1[j] = 0.f; s2[j] = 0.f; }
    for (int i = 0; i < 32; ++i) {
      v2f a, b;
      const float* pp = &poseS[((spA * 32 + i) * 4 + pA) * 4 + kq0];
      a.x = pp[0]; a.y = pp[1];
      const float* tp = tbase + (size_t)i * Oc * 16;
      b.x = tp[0] * msk; b.y = tp[4] * msk;
      v8f c = {0.f, 0.f, 0.f, 0.f, 0.f, 0.f, 0.f, 0.f};
      v8f d = __builtin_amdgcn_wmma_f32_16x16x4_f32(false, a, false, b, (short)0, c, false, false);
      for (int j = 0; j < 8; ++j) {
        float v = fmaxf(d[j], 0.f);
        s1[j] += v; s2[j] += v * v;
      }
    }
    for (int j = 0; j < 8; ++j) {
      int m2 = j + 8 * hi;
      int sp2 = m2 >> 2, p2 = m2 & 3;
      int idx = (sp2 * 32 + o) * 16 + p2 * 4 + rr;
      if (o < Oc) {
        meansS[idx] = s1[j];
        varS[idx] = s2[j] + 30.f * s1[j] * s1[j];
      } else {
        meansS[idx] = 0.f;
        varS[idx] = 1.f;
      }
    }
  }
  __syncthreads();

  // coeff_inv and cb = sum_k (beta_v - 0.5*log var); A = sum_i act
  const float TWO_PI = 6.283185307179586f;
  const int Opad = Ntiles << 2;
  for (int t = tid; t < 4 * Opad; t += 128) {
    int sp = t / Opad, o = t % Opad;
    float prod = 1.f, cb = 0.f;
    for (int k = 0; k < 16; ++k) {
      float v = varS[(sp * 32 + o) * 16 + k];
      prod *= TWO_PI * v;
      float bv = (o < Oc) ? beta_v[o * 16 + k] : 0.f;
      cb += bv - 0.5f * logf(v);
    }
    cinvS[sp * 32 + o] = sqrtf(prod);
    cbS[sp * 32 + o] = cb;
  }
  if (tid < 4) {
    float s = 0.f;
    for (int i = 0; i < 32; ++i) s += actS[tid * 32 + i];
    asumS[tid] = s;
  }
  __syncthreads();

  // -------- pass 2: G[o] = sum_i agr, H[o] = sum_i act_i * agr --------
  for (int ot = wid; ot < Ntiles; ot += 4) {
    int obase = ot << 2;
    int o = obase + ol;
    int oclamp = (o < Oc) ? o : 0;
    float msk = (o < Oc) ? 1.f : 0.f;
    const float* tbase = T + ((size_t)oclamp * 4 + kq0) * 4 + rr;
    float ci0 = cinvS[(2 * hi) * 32 + o];
    float ci1 = cinvS[(2 * hi + 1) * 32 + o];
    float g0 = 0.f, g1 = 0.f, h0 = 0.f, h1 = 0.f;
    for (int i = 0; i < 32; ++i) {
      v2f a, b;
      const float* pp = &poseS[((spA * 32 + i) * 4 + pA) * 4 + kq0];
      a.x = pp[0]; a.y = pp[1];
      const float* tp = tbase + (size_t)i * Oc * 16;
      b.x = tp[0] * msk; b.y = tp[4] * msk;
      v8f c = {0.f, 0.f, 0.f, 0.f, 0.f, 0.f, 0.f, 0.f};
      v8f d = __builtin_amdgcn_wmma_f32_16x16x4_f32(false, a, false, b, (short)0, c, false, false);
      float e0 = 0.f, e1 = 0.f;
      for (int j = 0; j < 8; ++j) {
        int m2 = j + 8 * hi;
        int sp2 = m2 >> 2, p2 = m2 & 3;
        int idx = (sp2 * 32 + o) * 16 + p2 * 4 + rr;
        float v = fmaxf(d[j], 0.f);
        float dv = v - meansS[idx];
        float contrib = dv * dv / (2.f * varS[idx]);
        if (j < 4) e0 += contrib; else e1 += contrib;
      }
      // reduce over rr lanes (bits 0,1 of lane)
      e0 += __shfl_xor(e0, 1, 32); e0 += __shfl_xor(e0, 2, 32);
      e1 += __shfl_xor(e1, 1, 32); e1 += __shfl_xor(e1, 2, 32);
      float ag0 = expf(-e0) / ci0;
      float ag1 = expf(-e1) / ci1;
      g0 += ag0; h0 += actS[(2 * hi) * 32 + i] * ag0;
      g1 += ag1; h1 += actS[(2 * hi + 1) * 32 + i] * ag1;
    }
    if (rr == 0 && o < Oc) {
      GS[(2 * hi) * 32 + o] = g0; HS[(2 * hi) * 32 + o] = h0;
      GS[(2 * hi + 1) * 32 + o] = g1; HS[(2 * hi + 1) * 32 + o] = h1;
    }
  }
  __syncthreads();

  // -------- routing recurrence (3 iters, closed form) + outputs --------
  for (int t = tid; t < 4 * Oc; t += 128) {
    int sp = t / Oc, o = t % Oc;
    float As = asumS[sp];
    float cb = cbS[sp * 32 + o];
    float G = GS[sp * 32 + o];
    float Hv = HS[sp * 32 + o];
    float ba = beta_a[o];
    float rsum = As / (float)Oc;
    float actout = 0.f;
    for (int it = 0; it < 3; ++it) {
      float z = (float)(it + 1) * (ba - rsum * cb);
      actout = 1.f / (1.f + expf(-z));
      if (it < 2) {
        float den = 32.f * EPSR + actout * G;
        rsum = (EPSR * As + actout * Hv) / den;
      }
    }
    act_out[(size_t)(pos0 + sp) * Oc + o] = actout;
  }
  for (int t = tid; t < 4 * Oc * 16; t += 128) {
    int sp = t / (Oc * 16);
    int rem = t % (Oc * 16);
    int o = rem >> 4;
    pose_out[(size_t)(pos0 + sp) * Oc * 16 + rem] = meansS[(sp * 32 + o) * 16 + (rem & 15)];
  }
}

extern "C" void kernel_launch(void* const* d_in, const int* in_sizes, int n_in,
                              void* d_out, int out_size, void* d_ws, size_t ws_size,
                              hipStream_t stream) {
  const float* x   = (const float*)d_in[0];
  const float* c1w = (const float*)d_in[1];
  const float* c1b = (const float*)d_in[2];
  const float* pw  = (const float*)d_in[3];
  const float* pb  = (const float*)d_in[4];
  const float* t1  = (const float*)d_in[5];
  const float* bv1 = (const float*)d_in[6];
  const float* ba1 = (const float*)d_in[7];
  const float* t2  = (const float*)d_in[8];
  const float* bv2 = (const float*)d_in[9];
  const float* ba2 = (const float*)d_in[10];
  const float* t3  = (const float*)d_in[11];
  const float* bv3 = (const float*)d_in[12];
  const float* ba3 = (const float*)d_in[13];
  float* out = (float*)d_out;

  float* ws = (float*)d_ws;
  float* y     = ws;                  // 50176
  float* poseA = y + 50176;           // 802816
  float* actA  = poseA + 802816;      // 50176
  float* poseB = actA + 50176;        // 802816
  float* actB  = poseB + 802816;      // 50176

  // conv1: 8*32*14*14 = 50176 threads
  caps_conv1_kernel<<<196, 256, 0, stream>>>(x, c1w, c1b, y);
  // pconv + reshape: 802816 + 50176 = 852992 threads
  caps_pconv_kernel<<<3332, 256, 0, stream>>>(y, pw, pb, poseA, actA);
  // capsule layers: 1568 positions / 4 per block = 392 blocks
  caps_layer_kernel<<<392, 128, 0, stream>>>(poseA, actA, t1, bv1, ba1, poseB, actB, 32);
  caps_layer_kernel<<<392, 128, 0, stream>>>(poseB, actB, t2, bv2, ba2, poseA, actA, 32);
  // layer 3: pose output unused; dump into poseB scratch, act -> d_out
  caps_layer_kernel<<<392, 128, 0, stream>>>(poseA, actA, t3, bv3, ba3, poseB, out, 10);
}